// HierarchicalStaticNeuralTexture_78159814853112
// MI455X (gfx1250) — compile-verified
//
#include <hip/hip_runtime.h>

// ---------------------------------------------------------------------------
// HierarchicalStaticNeuralTexture — MI455X (gfx1250)
//
// Phase 1 (repack): [16][2048][1024] channel-major fp32 -> compact channel-last
//   packed[texel][16] (64B per texel, 64B aligned), levels packed back-to-back:
//   level bases (texels): 0, 1024^2, +512^2, +256^2  (total 1,392,640 texels,
//   ~89 MB -> L2-resident on the 192 MB L2).
// Phase 2 (sample): 1 thread / pixel, 4 levels x 4 taps, each tap = 4x
//   global_load_b128 of a fully-used 64B texel record. All 16 tap addresses are
//   computed first and prefetched (global_prefetch_b8) before the blend chain.
// ---------------------------------------------------------------------------

namespace {
constexpr int IMG_PIX   = 1024 * 1024;       // pixels per channel plane
constexpr int CH_STRIDE = 2048 * 1024;       // per-channel plane in `data`
constexpr int N_TEXELS  = 1392640;           // 1024^2 + 512^2 + 256^2 + 128^2
constexpr size_t PACKED_BYTES = (size_t)N_TEXELS * 16u * sizeof(float);
}

// ---------------- Phase 1: channel-major -> channel-last repack -------------
__global__ __launch_bounds__(256) void hsnt_repack(const float* __restrict__ data,
                                                   float4* __restrict__ packed) {
  const int t = blockIdx.x * 256 + threadIdx.x;   // exact grid, no tail
  int w, offY, base, sh;
  if (t < 1048576)      { w = 1024; offY = 0;    base = 0;       sh = 10; }
  else if (t < 1310720) { w = 512;  offY = 1024; base = 1048576; sh = 9;  }
  else if (t < 1376256) { w = 256;  offY = 1536; base = 1310720; sh = 8;  }
  else                  { w = 128;  offY = 1792; base = 1376256; sh = 7;  }
  const int r = t - base;
  const int y = r >> sh;
  const int x = r & (w - 1);
  const float* __restrict__ s = data + (size_t)(offY + y) * 1024 + x;

  float4 v0, v1, v2, v3;
  v0.x = s[ 0 * CH_STRIDE]; v0.y = s[ 1 * CH_STRIDE];
  v0.z = s[ 2 * CH_STRIDE]; v0.w = s[ 3 * CH_STRIDE];
  v1.x = s[ 4 * CH_STRIDE]; v1.y = s[ 5 * CH_STRIDE];
  v1.z = s[ 6 * CH_STRIDE]; v1.w = s[ 7 * CH_STRIDE];
  v2.x = s[ 8 * CH_STRIDE]; v2.y = s[ 9 * CH_STRIDE];
  v2.z = s[10 * CH_STRIDE]; v2.w = s[11 * CH_STRIDE];
  v3.x = s[12 * CH_STRIDE]; v3.y = s[13 * CH_STRIDE];
  v3.z = s[14 * CH_STRIDE]; v3.w = s[15 * CH_STRIDE];

  float4* __restrict__ d = packed + (size_t)t * 4;
  d[0] = v0; d[1] = v1; d[2] = v2; d[3] = v3;   // 4x global_store_b128, 64B/thr
}

// ---------------- Phase 2: packed bilinear sample ---------------------------
__device__ __forceinline__ void fma4(float4& a, const float4& t, float w) {
  a.x = fmaf(t.x, w, a.x);
  a.y = fmaf(t.y, w, a.y);
  a.z = fmaf(t.z, w, a.z);
  a.w = fmaf(t.w, w, a.w);
}

__global__ __launch_bounds__(256) void hsnt_sample_packed(const float* __restrict__ uv,
                                                          const float4* __restrict__ packed,
                                                          float* __restrict__ out) {
  const int p  = blockIdx.x * 256 + threadIdx.x;  // exact grid
  const float gx = uv[p];
  const float gy = uv[IMG_PIX + p];

  int   idx[4][4];
  float wt[4][4];
#pragma unroll
  for (int L = 0; L < 4; ++L) {
    const int w    = 1024 >> L;
    const int base = (L == 0) ? 0 : (L == 1) ? 1048576 : (L == 2) ? 1310720 : 1376256;
    const float fw = (float)w;
    float xs = ((gx + 1.0f) * fw - 1.0f) * 0.5f;
    float ys = ((gy + 1.0f) * fw - 1.0f) * 0.5f;
    xs = fminf(fmaxf(xs, 0.0f), fw - 1.0f);
    ys = fminf(fmaxf(ys, 0.0f), fw - 1.0f);
    const float x0f = floorf(xs), y0f = floorf(ys);
    const float wx = xs - x0f,    wy = ys - y0f;
    const int x0 = (int)x0f,      y0 = (int)y0f;
    const int x1 = min(x0 + 1, w - 1), y1 = min(y0 + 1, w - 1);
    const int r0 = base + y0 * w;
    const int r1 = base + y1 * w;
    idx[L][0] = r0 + x0;  wt[L][0] = (1.0f - wy) * (1.0f - wx);
    idx[L][1] = r0 + x1;  wt[L][1] = (1.0f - wy) * wx;
    idx[L][2] = r1 + x0;  wt[L][2] = wy * (1.0f - wx);
    idx[L][3] = r1 + x1;  wt[L][3] = wy * wx;
  }

  // gfx1250 global_prefetch_b8: warm every tap line before the dependent chain
#pragma unroll
  for (int L = 0; L < 4; ++L)
#pragma unroll
    for (int k = 0; k < 4; ++k)
      __builtin_prefetch((const void*)(packed + (size_t)idx[L][k] * 4), 0, 3);

  float4 a0 = {0.f, 0.f, 0.f, 0.f};
  float4 a1 = {0.f, 0.f, 0.f, 0.f};
  float4 a2 = {0.f, 0.f, 0.f, 0.f};
  float4 a3 = {0.f, 0.f, 0.f, 0.f};
#pragma unroll
  for (int L = 0; L < 4; ++L) {
#pragma unroll
    for (int k = 0; k < 4; ++k) {
      const float4* __restrict__ t = packed + (size_t)idx[L][k] * 4;
      const float  w = wt[L][k];
      const float4 t0 = t[0], t1 = t[1], t2 = t[2], t3 = t[3]; // 4x b128 gather
      fma4(a0, t0, w); fma4(a1, t1, w); fma4(a2, t2, w); fma4(a3, t3, w);
    }
  }

  float* __restrict__ o = out + p;   // per-channel stores coalesce across lanes
  o[ 0 * IMG_PIX] = a0.x;  o[ 1 * IMG_PIX] = a0.y;
  o[ 2 * IMG_PIX] = a0.z;  o[ 3 * IMG_PIX] = a0.w;
  o[ 4 * IMG_PIX] = a1.x;  o[ 5 * IMG_PIX] = a1.y;
  o[ 6 * IMG_PIX] = a1.z;  o[ 7 * IMG_PIX] = a1.w;
  o[ 8 * IMG_PIX] = a2.x;  o[ 9 * IMG_PIX] = a2.y;
  o[10 * IMG_PIX] = a2.z;  o[11 * IMG_PIX] = a2.w;
  o[12 * IMG_PIX] = a3.x;  o[13 * IMG_PIX] = a3.y;
  o[14 * IMG_PIX] = a3.z;  o[15 * IMG_PIX] = a3.w;
}

// -------- Fallback: direct gather from [C,H,W] if workspace is too small ----
__global__ __launch_bounds__(256) void hsnt_sample_direct(const float* __restrict__ uv,
                                                          const float* __restrict__ data,
                                                          float* __restrict__ out) {
  const int p  = blockIdx.x * 256 + threadIdx.x;
  const float gx = uv[p];
  const float gy = uv[IMG_PIX + p];

  float acc[16];
#pragma unroll
  for (int c = 0; c < 16; ++c) acc[c] = 0.0f;

#pragma unroll
  for (int L = 0; L < 4; ++L) {
    const int w    = 1024 >> L;
    const int offY = (L == 0) ? 0 : (L == 1) ? 1024 : (L == 2) ? 1536 : 1792;
    const float fw = (float)w;
    float xs = ((gx + 1.0f) * fw - 1.0f) * 0.5f;
    float ys = ((gy + 1.0f) * fw - 1.0f) * 0.5f;
    xs = fminf(fmaxf(xs, 0.0f), fw - 1.0f);
    ys = fminf(fmaxf(ys, 0.0f), fw - 1.0f);
    const float x0f = floorf(xs), y0f = floorf(ys);
    const float wx = xs - x0f,    wy = ys - y0f;
    const int x0 = (int)x0f,      y0 = (int)y0f;
    const int x1 = min(x0 + 1, w - 1), y1 = min(y0 + 1, w - 1);

    const int o00 = (offY + y0) * 1024 + x0;
    const int o01 = (offY + y0) * 1024 + x1;
    const int o10 = (offY + y1) * 1024 + x0;
    const int o11 = (offY + y1) * 1024 + x1;
    const float w00 = (1.0f - wy) * (1.0f - wx);
    const float w01 = (1.0f - wy) * wx;
    const float w10 = wy * (1.0f - wx);
    const float w11 = wy * wx;
#pragma unroll
    for (int c = 0; c < 16; ++c) {
      const float* __restrict__ pl = data + (size_t)c * CH_STRIDE;
      float v = pl[o00] * w00;
      v = fmaf(pl[o01], w01, v);
      v = fmaf(pl[o10], w10, v);
      v = fmaf(pl[o11], w11, v);
      acc[c] += v;
    }
  }
#pragma unroll
  for (int c = 0; c < 16; ++c) out[(size_t)c * IMG_PIX + p] = acc[c];
}

// ---------------------------------------------------------------------------
extern "C" void kernel_launch(void* const* d_in, const int* in_sizes, int n_in,
                              void* d_out, int out_size, void* d_ws, size_t ws_size,
                              hipStream_t stream) {
  (void)in_sizes; (void)n_in; (void)out_size;
  const float* uv   = (const float*)d_in[0];   // [1,2,1024,1024]
  const float* data = (const float*)d_in[1];   // [1,16,2048,1024]
  float*       out  = (float*)d_out;           // [1,16,1024,1024]

  if (d_ws != nullptr && ws_size >= PACKED_BYTES) {
    float4* packed = (float4*)d_ws;
    hsnt_repack<<<N_TEXELS / 256, 256, 0, stream>>>(data, packed);
    hsnt_sample_packed<<<IMG_PIX / 256, 256, 0, stream>>>(uv, packed, out);
  } else {
    hsnt_sample_direct<<<IMG_PIX / 256, 256, 0, stream>>>(uv, data, out);
  }
}